// GraphConvolution_54331336295083
// MI455X (gfx1250) — compile-verified
//
#include <hip/hip_runtime.h>
#include <hip/hip_bf16.h>

// GCN layer: support = input @ weight ; out = relu(adj @ support)
// N = 16384, IN_F = 128, OUT_F = 32, all fp32.
//
// adj (1 GiB fp32) dominates: HBM-bandwidth-bound (~16 FLOP/byte).
// Strategy: fp32 WMMA (V_WMMA_F32_16X16X4_F32) on CDNA5 so matrix math
// overlaps the adj stream; support (2 MB) lives in L2 across all blocks.

typedef float v2f __attribute__((ext_vector_type(2)));
typedef float v8f __attribute__((ext_vector_type(8)));

#define GCN_N     16384
#define GCN_INF   128
#define GCN_OUTF  32

// ---------------------------------------------------------------------------
// Kernel 1: support[N, 32] = input[N, 128] @ weight[128, 32]
// One wave per output row: lanes share the input row, lanes = out columns.
// ---------------------------------------------------------------------------
__global__ void __launch_bounds__(256)
gcn_support_kernel(const float* __restrict__ input,
                   const float* __restrict__ weight,
                   float* __restrict__ support)
{
    const int t   = blockIdx.x * 256 + threadIdx.x;
    const int row = t >> 5;       // all 32 lanes of a wave -> same row
    const int col = t & 31;       // lane -> output column (coalesced weight)

    const float* __restrict__ irow = input + (size_t)row * GCN_INF;

    float sum = 0.0f;
#pragma unroll 8
    for (int j = 0; j < GCN_INF; ++j) {
        sum = fmaf(irow[j], weight[j * GCN_OUTF + col], sum);
    }
    support[(size_t)row * GCN_OUTF + col] = sum;
}

// ---------------------------------------------------------------------------
// Kernel 2: out = relu(adj @ support), one block per 16-row slab of adj.
// 8 wave32s per block, each wave owns an interleaved K-slice (stride 32),
// accumulates two 16x16 f32 tiles with V_WMMA_F32_16X16X4_F32, then the
// waves reduce through LDS (ds_add_f32) and ReLU+store.
//
// fp32 A 16x4 layout (ISA 7.12.2): lane<16 holds K=0(v0),K=1(v1) of row M=lane;
// lane>=16 holds K=2,K=3 of row M=lane-16.  B 4x16 mirrors it: v0/v1 hold rows
// K=0/K=1 (lanes 0-15, N=lane) and K=2/K=3 (lanes 16-31).
// C/D: VGPR r holds (M = r + 8*(lane>>4), N = lane&15).
// ---------------------------------------------------------------------------
__global__ void __launch_bounds__(256)
gcn_aggregate_kernel(const float* __restrict__ adj,
                     const float* __restrict__ support,
                     float* __restrict__ out)
{
    __shared__ float acc[16 * GCN_OUTF];   // 512 floats = 2 KB

    const int tid  = threadIdx.x;
    const int lane = tid & 31;
    const int wave = tid >> 5;
    const int rb   = blockIdx.x;           // rows rb*16 .. rb*16+15

    // zero the block accumulator
    acc[tid]       = 0.0f;
    acc[tid + 256] = 0.0f;
    __syncthreads();

    const int m      = lane & 15;            // tile row (A) / tile col (B,C)
    const int khalf  = (lane >> 4) << 1;     // 0 or 2: this lane's K sub-offset
    const int row    = rb * 16 + m;          // adj row this lane streams

    v8f c0 = {};                             // output cols 0..15
    v8f c1 = {};                             // output cols 16..31

    // Wave k-start interleaved: waves 0..7 cover k, k+4, ..., k+28 of each
    // 32-wide strip -> the workgroup consumes whole adj cachelines together.
    const float* __restrict__ ap =
        adj + (size_t)row * GCN_N + (size_t)(wave * 4 + khalf);
    const float* __restrict__ bp =
        support + (size_t)(wave * 4 + khalf) * GCN_OUTF + m;

#pragma unroll 4
    for (int it = 0; it < GCN_N / 32; ++it) {
        // A tile 16x4 fp32: 8 bytes per lane (K pair for this half-wave)
        v2f a = *(const v2f*)ap;

        // B tiles 4x16 fp32: rows khalf, khalf+1 for cols 0..15 and 16..31
        v2f b0 = { bp[0],  bp[GCN_OUTF] };
        v2f b1 = { bp[16], bp[GCN_OUTF + 16] };

        c0 = __builtin_amdgcn_wmma_f32_16x16x4_f32(
                 false, a, false, b0, (short)0, c0, false, false);
        c1 = __builtin_amdgcn_wmma_f32_16x16x4_f32(
                 false, a, false, b1, (short)0, c1, false, false);

        ap += 32;                  // next 32-wide K strip of this adj row
        bp += 32 * GCN_OUTF;       // matching support rows
    }

    // Cross-wave reduction into LDS (ds_add_f32).
    const int mbase = (lane >> 4) << 3;      // 0 or 8
#pragma unroll
    for (int r = 0; r < 8; ++r) {
        const int mm = mbase + r;
        atomicAdd(&acc[mm * GCN_OUTF + m],      c0[r]);
        atomicAdd(&acc[mm * GCN_OUTF + m + 16], c1[r]);
    }
    __syncthreads();

    // ReLU + coalesced store of the 16x32 slab.
    float* __restrict__ orow = out + (size_t)rb * 16 * GCN_OUTF;
    float v0 = acc[tid];
    float v1 = acc[tid + 256];
    orow[tid]       = v0 > 0.0f ? v0 : 0.0f;
    orow[tid + 256] = v1 > 0.0f ? v1 : 0.0f;
}

// ---------------------------------------------------------------------------
// Entry point
// ---------------------------------------------------------------------------
extern "C" void kernel_launch(void* const* d_in, const int* in_sizes, int n_in,
                              void* d_out, int out_size, void* d_ws, size_t ws_size,
                              hipStream_t stream) {
    const float* input  = (const float*)d_in[0];   // [16384, 128]
    const float* adj    = (const float*)d_in[1];   // [16384, 16384]
    const float* weight = (const float*)d_in[2];   // [128, 32]
    float*       out    = (float*)d_out;           // [16384, 32]
    float*       support = (float*)d_ws;           // [16384, 32] scratch (2 MB)

    // Kernel 1: 16384 rows, one wave each -> 16384*32 threads.
    gcn_support_kernel<<<(GCN_N * GCN_OUTF) / 256, 256, 0, stream>>>(
        input, weight, support);

    // Kernel 2: one block per 16-row slab.
    gcn_aggregate_kernel<<<GCN_N / 16, 256, 0, stream>>>(
        adj, support, out);
}